// VanillaKoopmanOperator_32109175505772
// MI455X (gfx1250) — compile-verified
//
#include <hip/hip_runtime.h>

// Koopman power iteration: z_{t+1} = z_t @ W, emit all T states.
// B=2048, K=256, T=256, fp32 (matches reference numerics).
//
// v3 strategy (MI455X / gfx1250):
//  - 128 WGs x 256 threads (8 waves). Each WG owns 16 rows of z; each wave
//    owns a 32-column slice (two 16x16 tiles).
//  - W lives entirely in VGPRs: each wave's B-operand set is invariant across
//    all 256 steps = 64 K-chunks x 2 tiles x v2f = 256 VGPRs, loaded once
//    from global (L2-resident). Steady state does zero W traffic.
//  - z state ping-pongs between two LDS buffers (stride 268 floats -> the 16
//    A-rows map to 16 distinct bank groups); one barrier per step.
//  - K-reduction split into low/high halves -> 4 independent WMMA accumulate
//    chains (dep distance 4) so issue never waits on accumulate forwarding.
//  - Inner loop fully unrolled: 32x ds_load_2addr_b64 (A-frags, pairs 512B
//    apart merge into one instr) + 128x v_wmma_f32_16x16x4_f32.
//  - Output is write-once: non-temporal stores, keep the 192MB L2 for W/z.

typedef __attribute__((ext_vector_type(2))) float v2f;
typedef __attribute__((ext_vector_type(4))) float v4f;
typedef __attribute__((ext_vector_type(8))) float v8f;

#define KDIM   256
#define TSTEPS 256
#define ZSTRIDE 268   // floats; row = 1072 B (16B aligned); banks 12m mod 64 distinct

__global__ __launch_bounds__(256, 1)
void koopman_wmma_f32(const float* __restrict__ x,
                      const float* __restrict__ W,
                      float* __restrict__ out) {
  __shared__ float lz[2][16 * ZSTRIDE];   // ping-pong z state, ~34 KB

  const int tid   = threadIdx.x;        // 0..255
  const int wave  = tid >> 5;           // 0..7
  const int lane  = tid & 31;
  const int lrow  = lane & 15;          // M for A-frag; N for B/D frags
  const int half  = lane >> 4;          // selects K pair (A/B) and M+8 (D)
  const int mbase = blockIdx.x << 4;    // this WG's 16-row base in B
  const int n0    = wave << 5;          // this wave's 32-column base

  // ---- One-time: hoist this wave's entire B-operand set into VGPRs ----
  // bfr[kc4][ti] = { W[kp][n], W[kp+1][n] },  kp = 4*kc4 + 2*half,
  // n = n0 + 16*ti + lrow.  64 x 2 x v2f = 256 VGPRs, held across the t-loop.
  v2f bfr[64][2];
#pragma unroll
  for (int kc4 = 0; kc4 < 64; ++kc4) {
    const int kp = (kc4 << 2) + (half << 1);
#pragma unroll
    for (int ti = 0; ti < 2; ++ti) {
      const int n = n0 + (ti << 4) + lrow;
      v2f b;
      b.x = W[kp * KDIM + n];
      b.y = W[(kp + 1) * KDIM + n];
      bfr[kc4][ti] = b;
    }
  }

  // ---- Initial state z0 = x[mbase : mbase+16, :] into buffer 0 ----
  for (int i = tid; i < 16 * 64; i += 256) {
    const int r = i >> 6;
    const int c = (i & 63) << 2;
    v4f v = *(const v4f*)(x + (size_t)(mbase + r) * KDIM + c);
    *(v4f*)(&lz[0][r * ZSTRIDE + c]) = v;
  }
  __syncthreads();

  int p = 0;
  for (int t = 0; t < TSTEPS; ++t) {
    const float* zb = &lz[p][0];
    float*       zn = &lz[p ^ 1][0];

    // 4 independent accumulate chains: (tile0, tile1) x (K-low, K-high).
    v8f acc0a = {}, acc0b = {};
    v8f acc1a = {}, acc1b = {};

    const float* arow = zb + lrow * ZSTRIDE + (half << 1);
#pragma unroll
    for (int kc4 = 0; kc4 < 32; ++kc4) {
      // A-frags 16x4: lane=M, VGPR0/1 = K = kp, kp+1. The two chunks are
      // 512 B apart -> one ds_load_2addr_b64 fetches both.
      v2f alo = *(const v2f*)(arow + (kc4 << 2));
      v2f ahi = *(const v2f*)(arow + ((kc4 + 32) << 2));
      acc0a = __builtin_amdgcn_wmma_f32_16x16x4_f32(
          false, alo, false, bfr[kc4][0],      (short)0, acc0a, false, false);
      acc1a = __builtin_amdgcn_wmma_f32_16x16x4_f32(
          false, alo, false, bfr[kc4][1],      (short)0, acc1a, false, false);
      acc0b = __builtin_amdgcn_wmma_f32_16x16x4_f32(
          false, ahi, false, bfr[kc4 + 32][0], (short)0, acc0b, false, false);
      acc1b = __builtin_amdgcn_wmma_f32_16x16x4_f32(
          false, ahi, false, bfr[kc4 + 32][1], (short)0, acc1b, false, false);
    }

    // Combine K-halves.
    v8f acc0, acc1;
#pragma unroll
    for (int v = 0; v < 8; ++v) {
      acc0[v] = acc0a[v] + acc0b[v];
      acc1[v] = acc1a[v] + acc1b[v];
    }

    // D-layout (§7.12.2): VGPR v holds M=v (lanes 0-15) / M=v+8 (lanes 16-31),
    // N = lane%16. Write z_{t+1} into the OTHER LDS buffer (readers of buffer
    // p are this step; writers target p^1 -> no barrier before writes), and
    // stream the state to out[:, t, :] with non-temporal stores.
    const int c0 = n0 + lrow;
    const int c1 = n0 + 16 + lrow;
#pragma unroll
    for (int v = 0; v < 8; ++v) {
      const int row = v + (half << 3);
      zn[row * ZSTRIDE + c0] = acc0[v];
      zn[row * ZSTRIDE + c1] = acc1[v];
      const size_t go = ((size_t)(mbase + row) * TSTEPS + t) * (size_t)KDIM;
      __builtin_nontemporal_store(acc0[v], out + go + c0);
      __builtin_nontemporal_store(acc1[v], out + go + c1);
    }

    __syncthreads();  // publish z_{t+1} before anyone reads buffer p^1
    p ^= 1;
  }
}

extern "C" void kernel_launch(void* const* d_in, const int* in_sizes, int n_in,
                              void* d_out, int out_size, void* d_ws, size_t ws_size,
                              hipStream_t stream) {
  (void)in_sizes; (void)n_in; (void)out_size; (void)d_ws; (void)ws_size;
  const float* x = (const float*)d_in[0];   // [2048, 256]
  const float* W = (const float*)d_in[1];   // [256, 256]
  // d_in[2] is T (=256), compiled in as TSTEPS.
  float* out = (float*)d_out;               // [2048, 256, 256]

  koopman_wmma_f32<<<dim3(2048 / 16), dim3(256), 0, stream>>>(x, W, out);
}